// GaussQuadLoss_49538152792620
// MI455X (gfx1250) — compile-verified
//
#include <hip/hip_runtime.h>
#include <hip/hip_bf16.h>
#include <math.h>
#include <float.h>

#define NN   16384
#define DD   256
#define MM   1024
#define RRC  0.9f
#define TMX  128
#define ROWS 32
#define PANEL 64
#define NPANEL (MM / PANEL)

typedef __attribute__((ext_vector_type(2))) float v2f;
typedef __attribute__((ext_vector_type(8))) float v8f;

__device__ __forceinline__ float nsc_f(float cnt) {
  float c  = fmaxf(cnt, 1e-10f);
  float Nm = fmaxf(1.0f, floorf(c + 0.5f));
  float rt = 1.0f, s = 0.0f;
#pragma unroll 8
  for (int T = 1; T <= TMX; ++T) { rt *= RRC; s += rt * fmaxf(Nm - (float)T, 0.0f); }
  float temp = (1.0f + 2.0f * (s / Nm)) / Nm;
  return temp * Nm / c;
}

// ---- (1) segment reductions: counts[M], sums[M,D] ----
__global__ void k_segsum(const float* __restrict__ z, const float* __restrict__ w,
                         const int* __restrict__ labels,
                         float* __restrict__ counts, float* __restrict__ sums) {
  int n = blockIdx.x;            // one block per sample, blockDim = D
  int d = threadIdx.x;
  int l = labels[n];
  float wn = w[n];
  atomicAdd(&sums[(size_t)l * DD + d], wn * z[(size_t)n * DD + d]);
  if (d == 0) atomicAdd(&counts[l], wn);
}

// ---- (2) prior, log prior, C0 = -sum p log p  (one block of M threads) ----
__global__ void k_stats(const float* __restrict__ counts,
                        float* __restrict__ logpr, float* __restrict__ scal) {
  __shared__ float red[MM];
  __shared__ float totS;
  int t = threadIdx.x;
  float c = counts[t];
  red[t] = c; __syncthreads();
  for (int s = MM / 2; s > 0; s >>= 1) { if (t < s) red[t] += red[t + s]; __syncthreads(); }
  if (t == 0) totS = red[0];
  __syncthreads();
  float p  = c / totS;
  float lp = logf(p);
  logpr[t] = lp;
  red[t] = -p * lp; __syncthreads();
  for (int s = MM / 2; s > 0; s >>= 1) { if (t < s) red[t] += red[t + s]; __syncthreads(); }
  if (t == 0) scal[2] = red[0];          // C0
}

// ---- (3) per-class Bayes adapt -> Amat = mu*inv, Bmat = -0.5*inv, c0 ----
__global__ void k_adapt(const float* __restrict__ counts, const float* __restrict__ sums,
                        const float* __restrict__ cov_wc, const float* __restrict__ cov_ac,
                        const float* __restrict__ logpr,
                        float* __restrict__ Amat, float* __restrict__ Bmat,
                        float* __restrict__ c0v) {
  int m = blockIdx.x, d = threadIdx.x;
  __shared__ float nscS;
  __shared__ float red[DD];
  float cnt = fmaxf(counts[m], 1e-10f);
  if (d == 0) nscS = nsc_f(cnt);
  __syncthreads();
  float nsc = nscS;
  float mu  = sums[(size_t)m * DD + d] / cnt;
  float cm  = cov_wc[d] * nsc;
  float tp  = cov_ac[d] / (cov_ac[d] + cm);
  mu *= tp;
  float cov = tp * cm;
  float inv = 1.0f / (cov + 1.0f);       // + COV_TEST
  Amat[(size_t)m * DD + d] = mu * inv;
  Bmat[(size_t)m * DD + d] = -0.5f * inv;
  red[d] = 0.5f * (logf(inv) - mu * mu * inv);
  __syncthreads();
  for (int s = DD / 2; s > 0; s >>= 1) { if (d < s) red[d] += red[d + s]; __syncthreads(); }
  if (d == 0) c0v[m] = red[0] + logpr[m];
}

// ---- (4) per-sample leave-one-out log-likelihood ----
__global__ void k_loo(const float* __restrict__ z, const float* __restrict__ w,
                      const int* __restrict__ labels,
                      const float* __restrict__ counts, const float* __restrict__ sums,
                      const float* __restrict__ cov_wc, const float* __restrict__ cov_ac,
                      float* __restrict__ llloo) {
  int n = blockIdx.x, d = threadIdx.x;
  __shared__ float nscS;
  __shared__ float red[DD];
  int l = labels[n];
  float wn = w[n];
  float zv = z[(size_t)n * DD + d];
  float cnt = fmaxf(counts[l] - wn, 1e-10f);
  if (d == 0) nscS = nsc_f(cnt);
  __syncthreads();
  float nsc = nscS;
  float sn  = sums[(size_t)l * DD + d] - wn * zv;
  float mu  = sn / cnt;
  float cm  = cov_wc[d] * nsc;
  float tp  = cov_ac[d] / (cov_ac[d] + cm);
  mu *= tp;
  float covt = tp * cm + 1.0f;
  float diff = zv - mu;
  red[d] = logf(covt) + diff * diff / covt;
  __syncthreads();
  for (int s = DD / 2; s > 0; s >>= 1) { if (d < s) red[d] += red[d + s]; __syncthreads(); }
  if (d == 0) llloo[n] = -0.5f * red[0];
}

// ---- (5) fused WMMA GEMM (K = 2*D via X and X^2) + online softmax + NCE ----
__global__ void __launch_bounds__(256) k_score(
    const float* __restrict__ z, const float* __restrict__ w,
    const int* __restrict__ labels, const float* __restrict__ counts,
    const float* __restrict__ Amat, const float* __restrict__ Bmat,
    const float* __restrict__ c0v, const float* __restrict__ logpr,
    const float* __restrict__ llloo, float* __restrict__ scal) {
  __shared__ float Xt[ROWS][DD];        // 32 KB X tile
  __shared__ float Pbuf[ROWS][PANEL];   // 8 KB panel of LL
  __shared__ float s_m[ROWS], s_s[ROWS], s_best[ROWS], s_lab[ROWS];
  __shared__ int   s_bidx[ROWS];

  const int tid = threadIdx.x;
  const int gr0 = blockIdx.x * ROWS;

  for (int idx = tid; idx < ROWS * DD; idx += 256)
    Xt[idx / DD][idx % DD] = z[(size_t)(gr0 + idx / DD) * DD + (idx % DD)];

  if (tid < ROWS) {
    s_m[tid] = -FLT_MAX; s_s[tid] = 0.0f;
    s_best[tid] = -FLT_MAX; s_bidx[tid] = 0x7fffffff; s_lab[tid] = 0.0f;
  }
  __syncthreads();

  // 8 waves: 2 row-groups x 4 col-groups of 16x16 WMMA tiles
  const int wv   = tid >> 5;
  const int lane = tid & 31;
  const int rg   = wv >> 2;
  const int cg   = wv & 3;
  const int kh   = lane >> 4;       // half-wave selects K pair
  const int l16  = lane & 15;
  const int arow = rg * 16 + l16;   // A-matrix row (block-local)

  // softmax-update mapping: 8 threads per row, 8 cols each
  const int   ur    = tid >> 3;
  const int   uj    = tid & 7;
  const int   ugrow = gr0 + ur;
  const int   ulab  = labels[ugrow];
  const float uw    = w[ugrow];
  const bool  uvalid = (counts[ulab] - uw) > 1e-8f;
  const float usub   = llloo[ugrow] + logpr[ulab];

  for (int p = 0; p < NPANEL; ++p) {
    const int    mcol = p * PANEL + cg * 16 + l16;   // class = B column
    const size_t boff = (size_t)mcol * DD;
    v8f acc = {};
    // LL += X @ Amat^T
#pragma unroll 4
    for (int k = 0; k < DD; k += 4) {
      const int kk = k + 2 * kh;
      v2f a = *(const v2f*)&Xt[arow][kk];
      v2f b = *(const v2f*)(Amat + boff + kk);
      acc = __builtin_amdgcn_wmma_f32_16x16x4_f32(false, a, false, b,
                                                  (short)0, acc, false, false);
    }
    // LL += X^2 @ (-0.5*inv)^T
#pragma unroll 4
    for (int k = 0; k < DD; k += 4) {
      const int kk = k + 2 * kh;
      v2f a = *(const v2f*)&Xt[arow][kk];
      a[0] *= a[0]; a[1] *= a[1];
      v2f b = *(const v2f*)(Bmat + boff + kk);
      acc = __builtin_amdgcn_wmma_f32_16x16x4_f32(false, a, false, b,
                                                  (short)0, acc, false, false);
    }
    const float cc0  = c0v[mcol];
    const int   prow = rg * 16 + 8 * kh;
    const int   pcol = cg * 16 + l16;
#pragma unroll
    for (int i = 0; i < 8; ++i)
      Pbuf[prow + i][pcol] = acc[i] + cc0;
    __syncthreads();

    // online softmax / argmax update over this 64-wide panel
    float v[8];
    float lm = -FLT_MAX;
#pragma unroll
    for (int i = 0; i < 8; ++i) {
      int jc = uj * 8 + i;
      float vv = Pbuf[ur][jc];
      int mc = p * PANEL + jc;
      if (mc == ulab) {                 // LOO substitution before max/sum
        if (uvalid) vv = usub;
        s_lab[ur] = vv;
      }
      v[i] = vv;
      lm = fmaxf(lm, vv);
    }
    float bval = -FLT_MAX; int bidx = 0x7fffffff;
#pragma unroll
    for (int i = 0; i < 8; ++i) {
      int mc = p * PANEL + uj * 8 + i;
      if (v[i] > bval) { bval = v[i]; bidx = mc; }
    }
    for (int off = 4; off > 0; off >>= 1) {
      lm = fmaxf(lm, __shfl_xor(lm, off, 8));
      float ov = __shfl_xor(bval, off, 8);
      int   oi = __shfl_xor(bidx, off, 8);
      if (ov > bval || (ov == bval && oi < bidx)) { bval = ov; bidx = oi; }
    }
    float ls = 0.0f;
#pragma unroll
    for (int i = 0; i < 8; ++i) ls += __expf(v[i] - lm);
    for (int off = 4; off > 0; off >>= 1) ls += __shfl_xor(ls, off, 8);

    if (uj == 0) {
      float m_old = s_m[ur];
      float m_new = fmaxf(m_old, lm);
      s_s[ur] = s_s[ur] * __expf(m_old - m_new) + ls * __expf(lm - m_new);
      s_m[ur] = m_new;
      if (bval > s_best[ur] || (bval == s_best[ur] && bidx < s_bidx[ur])) {
        s_best[ur] = bval; s_bidx[ur] = bidx;
      }
    }
    __syncthreads();
  }

  if (tid < ROWS) {
    float lse = s_m[tid] + logf(s_s[tid]);
    atomicAdd(&scal[0], lse - s_lab[tid]);          // sum of -log p(label)
    int lab = labels[gr0 + tid];
    if (s_bidx[tid] == lab) atomicAdd(&scal[1], 1.0f);
  }
}

// ---- (6) finish: loss = sum/(N*C0), acc = correct/N ----
__global__ void k_finish(const float* __restrict__ scal, float* __restrict__ out) {
  out[0] = scal[0] / ((float)NN * scal[2]);
  out[1] = scal[1] / (float)NN;
}

extern "C" void kernel_launch(void* const* d_in, const int* in_sizes, int n_in,
                              void* d_out, int out_size, void* d_ws, size_t ws_size,
                              hipStream_t stream) {
  (void)in_sizes; (void)n_in; (void)out_size; (void)ws_size;
  const float* z      = (const float*)d_in[0];
  const float* w      = (const float*)d_in[1];
  const float* cov_wc = (const float*)d_in[2];
  const float* cov_ac = (const float*)d_in[3];
  const int*   labels = (const int*)d_in[4];

  float* ws     = (float*)d_ws;
  float* counts = ws;                          // M
  float* sums   = counts + MM;                 // M*D
  float* Amat   = sums + (size_t)MM * DD;      // M*D
  float* Bmat   = Amat + (size_t)MM * DD;      // M*D
  float* c0v    = Bmat + (size_t)MM * DD;      // M
  float* logpr  = c0v + MM;                    // M
  float* llloo  = logpr + MM;                  // N
  float* scal   = llloo + NN;                  // 4: loss-sum, correct, C0
  float* out    = (float*)d_out;

  hipMemsetAsync(counts, 0, ((size_t)MM + (size_t)MM * DD) * sizeof(float), stream);
  hipMemsetAsync(scal, 0, 4 * sizeof(float), stream);

  k_segsum<<<NN, DD, 0, stream>>>(z, w, labels, counts, sums);
  k_stats <<<1, MM, 0, stream>>>(counts, logpr, scal);
  k_adapt <<<MM, DD, 0, stream>>>(counts, sums, cov_wc, cov_ac, logpr, Amat, Bmat, c0v);
  k_loo   <<<NN, DD, 0, stream>>>(z, w, labels, counts, sums, cov_wc, cov_ac, llloo);
  k_score <<<NN / ROWS, 256, 0, stream>>>(z, w, labels, counts, Amat, Bmat, c0v,
                                          logpr, llloo, scal);
  k_finish<<<1, 1, 0, stream>>>(scal, out);
}